// BodyModel_80504866996303
// MI455X (gfx1250) — compile-verified
//
#include <hip/hip_runtime.h>
#include <hip/hip_bf16.h>

// ---------------- problem constants (match reference) ----------------
#define NB_BATCH 512
#define NV       6890
#define NJ       24
#define NBETA    10
#define KP       207              // (J-1)*9
#define NCOL     (NV * 3)         // 20670
#define KTOT     (KP + NBETA + 1) // 218 augmented K
#define KPAD     224              // 14 chunks of 16

typedef __attribute__((ext_vector_type(2))) float v2f;
typedef __attribute__((ext_vector_type(8))) float v8f;

// ---------------- kernel 1: Rodrigues + pose_feature ----------------
__global__ __launch_bounds__(256) void rodrigues_kernel(
    const float* __restrict__ pose,   // B x 72
    float* __restrict__ rot,          // B x J x 9
    float* __restrict__ pf)           // B x 207
{
    int idx = blockIdx.x * blockDim.x + threadIdx.x;
    if (idx >= NB_BATCH * NJ) return;
    int b = idx / NJ, j = idx % NJ;
    float rx = pose[b * 72 + j * 3 + 0];
    float ry = pose[b * 72 + j * 3 + 1];
    float rz = pose[b * 72 + j * 3 + 2];
    float a2 = rx * rx + ry * ry + rz * rz + 1e-16f;
    float ang = sqrtf(a2);
    float s = __sinf(ang), c = __cosf(ang);
    float inv = 1.0f / ang;
    float x = rx * inv, y = ry * inv, z = rz * inv;
    float ic = 1.0f - c;
    float R[9];
    R[0] = c + ic * x * x;  R[1] = ic * x * y - s * z; R[2] = ic * x * z + s * y;
    R[3] = ic * x * y + s * z; R[4] = c + ic * y * y;  R[5] = ic * y * z - s * x;
    R[6] = ic * x * z - s * y; R[7] = ic * y * z + s * x; R[8] = c + ic * z * z;
#pragma unroll
    for (int i = 0; i < 9; ++i) rot[idx * 9 + i] = R[i];
    if (j >= 1) {
#pragma unroll
        for (int i = 0; i < 9; ++i) {
            float e = ((i == 0) | (i == 4) | (i == 8)) ? 1.0f : 0.0f;
            pf[b * KP + (j - 1) * 9 + i] = R[i] - e;
        }
    }
}

// ---------------- kernel 2: contract J_regressor with template/shapedirs ----
__global__ __launch_bounds__(256) void jreg_kernel(
    const float* __restrict__ Jreg,       // J x V
    const float* __restrict__ v_template, // V x 3
    const float* __restrict__ shapedirs,  // V x 3 x 10
    float* __restrict__ JregV,            // J*3
    float* __restrict__ JregS)            // J*3 x 10
{
    int jc = blockIdx.x;          // 0..71
    int j = jc / 3, c = jc % 3;
    int tid = threadIdx.x;
    float vals[11];
#pragma unroll
    for (int i = 0; i < 11; ++i) vals[i] = 0.0f;
    for (int v = tid; v < NV; v += 256) {
        float w = Jreg[j * NV + v];
        vals[0] += w * v_template[v * 3 + c];
#pragma unroll
        for (int k = 0; k < NBETA; ++k)
            vals[1 + k] += w * shapedirs[(v * 3 + c) * NBETA + k];
    }
    __shared__ float sd[256];
#pragma unroll
    for (int i = 0; i < 11; ++i) {
        sd[tid] = vals[i];
        __syncthreads();
        for (int s = 128; s > 0; s >>= 1) {
            if (tid < s) sd[tid] += sd[tid + s];
            __syncthreads();
        }
        if (tid == 0) {
            if (i == 0) JregV[jc] = sd[0];
            else        JregS[jc * NBETA + (i - 1)] = sd[0];
        }
        __syncthreads();
    }
}

// ---------------- kernel 3: per-batch kinematic chain -> A (3x4) ----------
__global__ __launch_bounds__(128) void chain_kernel(
    const float* __restrict__ betas,    // B x 10
    const float* __restrict__ rot,      // B x J x 9
    const float* __restrict__ JregV,    // J*3
    const float* __restrict__ JregS,    // J*3 x 10
    const int*   __restrict__ parents,  // J
    float* __restrict__ Amat)           // B x J x 12  ([R|t] rows)
{
    int b = blockIdx.x * blockDim.x + threadIdx.x;
    if (b >= NB_BATCH) return;
    float bt[NBETA];
#pragma unroll
    for (int k = 0; k < NBETA; ++k) bt[k] = betas[b * NBETA + k];

    float Jts[NJ][3];
    for (int j = 0; j < NJ; ++j)
        for (int c = 0; c < 3; ++c) {
            float s = JregV[j * 3 + c];
#pragma unroll
            for (int k = 0; k < NBETA; ++k) s += bt[k] * JregS[(j * 3 + c) * NBETA + k];
            Jts[j][c] = s;
        }

    float W[NJ][12];  // world [R|t], rows of 4
    for (int j = 0; j < NJ; ++j) {
        float R[9];
#pragma unroll
        for (int i = 0; i < 9; ++i) R[i] = rot[(b * NJ + j) * 9 + i];
        int p = (j == 0) ? 0 : parents[j];
        float rel[3];
#pragma unroll
        for (int c = 0; c < 3; ++c) rel[c] = Jts[j][c] - ((j == 0) ? 0.0f : Jts[p][c]);
        if (j == 0) {
#pragma unroll
            for (int r = 0; r < 3; ++r) {
#pragma unroll
                for (int c2 = 0; c2 < 3; ++c2) W[0][r * 4 + c2] = R[r * 3 + c2];
                W[0][r * 4 + 3] = rel[r];
            }
        } else {
#pragma unroll
            for (int r = 0; r < 3; ++r) {
#pragma unroll
                for (int c2 = 0; c2 < 3; ++c2)
                    W[j][r * 4 + c2] = W[p][r * 4 + 0] * R[0 * 3 + c2]
                                     + W[p][r * 4 + 1] * R[1 * 3 + c2]
                                     + W[p][r * 4 + 2] * R[2 * 3 + c2];
                W[j][r * 4 + 3] = W[p][r * 4 + 0] * rel[0]
                                + W[p][r * 4 + 1] * rel[1]
                                + W[p][r * 4 + 2] * rel[2]
                                + W[p][r * 4 + 3];
            }
        }
#pragma unroll
        for (int r = 0; r < 3; ++r) {
#pragma unroll
            for (int c2 = 0; c2 < 3; ++c2)
                Amat[(b * NJ + j) * 12 + r * 4 + c2] = W[j][r * 4 + c2];
            Amat[(b * NJ + j) * 12 + r * 4 + 3] =
                W[j][r * 4 + 3] - (W[j][r * 4 + 0] * Jts[j][0]
                                 + W[j][r * 4 + 1] * Jts[j][1]
                                 + W[j][r * 4 + 2] * Jts[j][2]);
        }
    }
}

// ---------------- kernel 4: augmented WMMA GEMM -> v_posed ----------------
// v_posed(512 x 20670) = [pf | betas | 1](512 x 218) @ [posedirs ; shapedirs^T ; v_template^T](218 x 20670)
// Branchy boundary logic lives ONLY in the LDS staging phase; the wmma loop is
// pure ds_load_b64 + v_wmma.
__device__ __forceinline__ float aug_a(int k, int m,
                                       const float* __restrict__ pf,
                                       const float* __restrict__ betas) {
    if (k < KP)          return pf[m * KP + k];
    if (k < KP + NBETA)  return betas[m * NBETA + (k - KP)];
    if (k == KP + NBETA) return 1.0f;
    return 0.0f;
}
__device__ __forceinline__ float aug_b(int k, int n,
                                       const float* __restrict__ posedirs,
                                       const float* __restrict__ shapedirs,
                                       const float* __restrict__ v_template) {
    if (n >= NCOL)       return 0.0f;
    if (k < KP)          return posedirs[k * NCOL + n];
    if (k < KP + NBETA)  return shapedirs[n * NBETA + (k - KP)];
    if (k == KP + NBETA) return v_template[n];
    return 0.0f;
}

#define TM 64   // block M tile (4 waves x 16)
#define TN 64   // block N tile
#define TK 16   // K chunk
#define AS_STRIDE 20   // padded LDS row stride for A (even -> 8B-aligned v2f, spreads banks)
// B stored k-pair interleaved: Bs[(k>>1)][n][k&1]  -> (b.x,b.y) is one aligned b64.
// Pair-row stride 160 floats: 160 % 64 == 32, so the upper half-wave (next pair row)
// hits banks 32..63 while lanes 0-15 hit 0..31 -> conflict-free ds_load_b64.
#define BP_STRIDE 160

__global__ __launch_bounds__(128) void pose_gemm_kernel(
    const float* __restrict__ pf,         // B x 207
    const float* __restrict__ betas,      // B x 10
    const float* __restrict__ posedirs,   // 207 x 20670
    const float* __restrict__ shapedirs,  // 20670 x 10 (flattened (V,3,10))
    const float* __restrict__ v_template, // 20670
    float* __restrict__ v_posed)          // 512 x 20670
{
    __shared__ __align__(16) float As[TM * AS_STRIDE];        // [m][k]
    __shared__ __align__(16) float Bs[(TK / 2) * BP_STRIDE];  // [k/2][n][k&1]

    const int tid  = threadIdx.x;
    const int wave = tid >> 5;       // 0..3
    const int lane = tid & 31;
    const int half = lane >> 4;      // 0 | 1
    const int l16  = lane & 15;

    const int m0 = blockIdx.x * TM;  // 8 blocks, exact
    const int n0 = blockIdx.y * TN;  // 323 blocks, last partial

    v8f acc[4];
#pragma unroll
    for (int t = 0; t < 4; ++t) acc[t] = (v8f){0.f,0.f,0.f,0.f,0.f,0.f,0.f,0.f};

    for (int kk = 0; kk < KPAD; kk += TK) {
        // ---- stage A chunk: 64 rows x 16 k ----
#pragma unroll
        for (int j = 0; j < (TM * TK) / 128; ++j) {
            int i = tid + j * 128;
            int r = i >> 4, c = i & 15;
            As[r * AS_STRIDE + c] = aug_a(kk + c, m0 + r, pf, betas);
        }
        // ---- stage B chunk: 16 k x 64 n, k-pair interleaved ----
#pragma unroll
        for (int j = 0; j < (TK * TN) / 128; ++j) {
            int i = tid + j * 128;
            int r = i >> 6, c = i & 63;
            Bs[(r >> 1) * BP_STRIDE + c * 2 + (r & 1)] =
                aug_b(kk + r, n0 + c, posedirs, shapedirs, v_template);
            // prefetch next chunk's posedirs row into cache (gfx1250 global_prefetch)
            int kn = kk + TK + r;
            if (kn < KP && (n0 + c) < NCOL)
                __builtin_prefetch(&posedirs[kn * NCOL + n0 + c], 0, 1);
        }
        __syncthreads();

        // ---- compute: pure LDS b64 + WMMA, branch-free ----
        const float* Arow = As + (wave * 16 + l16) * AS_STRIDE;
#pragma unroll
        for (int k0 = 0; k0 < TK; k0 += 4) {
            const int ks = k0 + half * 2;   // lanes 0-15: k0,k0+1 ; 16-31: k0+2,k0+3  (even)
            v2f a = *(const v2f*)(Arow + ks);
            const float* Brow = Bs + (ks >> 1) * BP_STRIDE;
#pragma unroll
            for (int t = 0; t < 4; ++t) {
                v2f b = *(const v2f*)(Brow + (t * 16 + l16) * 2);
                acc[t] = __builtin_amdgcn_wmma_f32_16x16x4_f32(
                    false, a, false, b, (short)0, acc[t], false, false);
            }
        }
        __syncthreads();
    }

    // ---- store: element e of v8f -> row (e + 8*half), col l16 within 16x16 tile
    const int mw = m0 + wave * 16;
#pragma unroll
    for (int t = 0; t < 4; ++t) {
        const int n = n0 + t * 16 + l16;
        if (n < NCOL) {
#pragma unroll
            for (int e = 0; e < 8; ++e) {
                const int row = mw + e + 8 * half;
                v_posed[(size_t)row * NCOL + n] = acc[t][e];
            }
        }
    }
}

// ---------------- kernel 5: skinning ----------------
__global__ __launch_bounds__(256) void skin_kernel(
    const float* __restrict__ Amat,     // B x J x 12
    const float* __restrict__ weights,  // V x J
    const float* __restrict__ v_posed,  // B x V*3
    const float* __restrict__ trans,    // B x 3
    float* __restrict__ out)            // B x V x 3
{
    __shared__ float sA[NJ * 12];
    const int b = blockIdx.y;
    for (int i = threadIdx.x; i < NJ * 12; i += 256) sA[i] = Amat[b * NJ * 12 + i];
    __syncthreads();
    const int v = blockIdx.x * 256 + threadIdx.x;
    if (v >= NV) return;

    float T[12];
#pragma unroll
    for (int i = 0; i < 12; ++i) T[i] = 0.0f;
#pragma unroll
    for (int j = 0; j < NJ; ++j) {
        const float w = weights[v * NJ + j];
#pragma unroll
        for (int i = 0; i < 12; ++i) T[i] += w * sA[j * 12 + i];
    }
    const size_t base = (size_t)b * NCOL + (size_t)v * 3;
    const float px = v_posed[base + 0];
    const float py = v_posed[base + 1];
    const float pz = v_posed[base + 2];
    const float tx = trans[b * 3 + 0], ty = trans[b * 3 + 1], tz = trans[b * 3 + 2];
    out[base + 0] = T[0] * px + T[1] * py + T[2]  * pz + T[3]  + tx;
    out[base + 1] = T[4] * px + T[5] * py + T[6]  * pz + T[7]  + ty;
    out[base + 2] = T[8] * px + T[9] * py + T[10] * pz + T[11] + tz;
}

// ---------------- launcher ----------------
extern "C" void kernel_launch(void* const* d_in, const int* in_sizes, int n_in,
                              void* d_out, int out_size, void* d_ws, size_t ws_size,
                              hipStream_t stream) {
    const float* pose       = (const float*)d_in[0];
    const float* betas      = (const float*)d_in[1];
    const float* trans      = (const float*)d_in[2];
    const float* v_template = (const float*)d_in[3];
    const float* shapedirs  = (const float*)d_in[4];
    const float* posedirs   = (const float*)d_in[5];
    const float* Jreg       = (const float*)d_in[6];
    const float* weights    = (const float*)d_in[7];
    const int*   parents    = (const int*)d_in[8];
    float* out = (float*)d_out;

    // workspace layout (floats): ~43.8 MB total
    float* ws      = (float*)d_ws;
    float* rot     = ws;                                  // B*J*9    = 110592
    float* pf      = rot   + NB_BATCH * NJ * 9;           // B*207    = 105984
    float* jregV   = pf    + NB_BATCH * KP;               // 72
    float* jregS   = jregV + NJ * 3;                      // 720
    float* Amat    = jregS + NJ * 3 * NBETA;              // B*J*12   = 147456
    float* v_posed = Amat  + NB_BATCH * NJ * 12;          // B*V*3    = 10583040

    // 1) Rodrigues + pose_feature
    rodrigues_kernel<<<(NB_BATCH * NJ + 255) / 256, 256, 0, stream>>>(pose, rot, pf);
    // 2) J_regressor precontractions
    jreg_kernel<<<NJ * 3, 256, 0, stream>>>(Jreg, v_template, shapedirs, jregV, jregS);
    // 3) kinematic chain -> A
    chain_kernel<<<(NB_BATCH + 127) / 128, 128, 0, stream>>>(betas, rot, jregV, jregS, parents, Amat);
    // 4) augmented WMMA GEMM -> v_posed (template + shape blend + pose blend fused)
    dim3 ggrid(NB_BATCH / TM, (NCOL + TN - 1) / TN);
    pose_gemm_kernel<<<ggrid, 128, 0, stream>>>(pf, betas, posedirs, shapedirs, v_template, v_posed);
    // 5) skinning
    dim3 sgrid((NV + 255) / 256, NB_BATCH);
    skin_kernel<<<sgrid, 256, 0, stream>>>(Amat, weights, v_posed, trans, out);
}